// mBERThidden4LossPosemb_77343771066430
// MI455X (gfx1250) — compile-verified
//
#include <hip/hip_runtime.h>
#include <math.h>

typedef __attribute__((ext_vector_type(2))) float v2f;
typedef __attribute__((ext_vector_type(8))) float v8f;

#define Bn   64
#define Ln   512
#define Hn   768
#define En   128
#define NROLE 64
#define FCIN 1920
#define BL   (Bn*Ln)          // 32768
#define BH   (Bn*Hn)          // 49152
#define BLL  ((size_t)Bn*Ln*Ln) // 16777216

__device__ __forceinline__ float wred(float v) {
  #pragma unroll
  for (int m = 16; m; m >>= 1) v += __shfl_xor(v, m, 32);
  return v;
}

__device__ __forceinline__ float logsig(float x) {
  // log(sigmoid(x)) = -log1p(exp(-|x|)) + min(x,0)
  float l = -log1pf(__expf(-fabsf(x)));
  return (x >= 0.f) ? l : x + l;
}

// ---- mask counts: one wave per batch ----------------------------------
__global__ __launch_bounds__(256) void k_maskcnt(const int* __restrict__ tm,
                                                 const int* __restrict__ am,
                                                 float* tcnt, float* acnt) {
  int b = blockIdx.x * 8 + (threadIdx.x >> 5);
  int lane = threadIdx.x & 31;
  float at = 0.f, aa = 0.f;
  for (int l = lane; l < Ln; l += 32) {
    at += (float)tm[b * Ln + l];
    aa += (float)am[b * Ln + l];
  }
  at = wred(at); aa = wred(aa);
  if (!lane) { tcnt[b] = at; acnt[b] = aa; }
}

// ---- masked means of transformer_output -------------------------------
__global__ __launch_bounds__(256) void k_masked_mean(const float* __restrict__ X,
                                                     const int* __restrict__ tm,
                                                     const int* __restrict__ am,
                                                     const float* tcnt, const float* acnt,
                                                     float* trig, float* arg) {
  int idx = blockIdx.x * 256 + threadIdx.x;
  if (idx >= BH) return;
  int b = idx / Hn, h = idx % Hn;
  const float* p = X + (size_t)b * Ln * Hn + h;
  float s1 = 0.f, s2 = 0.f;
  for (int l = 0; l < Ln; ++l) {
    float x = p[(size_t)l * Hn];
    s1 += x * (float)tm[b * Ln + l];
    s2 += x * (float)am[b * Ln + l];
  }
  trig[idx] = s1 / tcnt[b];
  arg[idx]  = s2 / acnt[b];
}

// ---- row L2 norms for both tensors (one wave per row) -----------------
__global__ __launch_bounds__(256) void k_rownorm(const float* __restrict__ V,
                                                 const float* __restrict__ Q,
                                                 float* vn, float* qn) {
  int wave = blockIdx.x * 8 + (threadIdx.x >> 5);   // 0..65535
  int lane = threadIdx.x & 31;
  const float* src = (wave < BL) ? V : Q;
  float* dst       = (wave < BL) ? vn : qn;
  int row = wave & (BL - 1);
  const float* p = src + (size_t)row * Hn;
  float acc = 0.f;
  for (int h = lane; h < Hn; h += 32) { float x = p[h]; acc += x * x; }
  acc = wred(acc);
  if (!lane) dst[row] = sqrtf(acc);
}

// ---- GEMM1 via V_WMMA_F32_16X16X4_F32, 2x2 register tiling ------------
// Each wave computes a 32x32 block of SIM[b] -> 1 fragment load per WMMA.
__global__ __launch_bounds__(256) void k_gemm_sim(const float* __restrict__ V,
                                                  const float* __restrict__ Q,
                                                  const float* __restrict__ vn,
                                                  const float* __restrict__ qn,
                                                  float* __restrict__ SIM) {
  const int wave = blockIdx.x * 8 + (threadIdx.x >> 5);  // 16384 waves
  const int lane = threadIdx.x & 31;
  const int b  = wave >> 8;          // 16x16 blocks of 32x32 per batch
  const int t  = wave & 255;
  const int i0 = (t >> 4) << 5;
  const int j0 = (t & 15) << 5;
  const int lm = lane & 15;
  const int lh = lane >> 4;

  // A fragment (16x4, MxK): lane lm holds K = 2*lh + {0,1}  -> float2
  // B fragment (4x16, KxN): lane lm (=N) holds K = 2*lh + {0,1} -> float2
  const float* pa0 = V + ((size_t)b * Ln + (i0 + lm)) * Hn + 2 * lh;
  const float* pa1 = pa0 + 16 * Hn;
  const float* pb0 = Q + ((size_t)b * Ln + (j0 + lm)) * Hn + 2 * lh;
  const float* pb1 = pb0 + 16 * Hn;

  v8f c00 = {}, c01 = {}, c10 = {}, c11 = {};
  #pragma unroll 2
  for (int k = 0; k < Hn; k += 4) {
    v2f a0 = *(const v2f*)(pa0 + k);
    v2f a1 = *(const v2f*)(pa1 + k);
    v2f b0 = *(const v2f*)(pb0 + k);
    v2f b1 = *(const v2f*)(pb1 + k);
    c00 = __builtin_amdgcn_wmma_f32_16x16x4_f32(false, a0, false, b0, (short)0, c00, false, false);
    c01 = __builtin_amdgcn_wmma_f32_16x16x4_f32(false, a0, false, b1, (short)0, c01, false, false);
    c10 = __builtin_amdgcn_wmma_f32_16x16x4_f32(false, a1, false, b0, (short)0, c10, false, false);
    c11 = __builtin_amdgcn_wmma_f32_16x16x4_f32(false, a1, false, b1, (short)0, c11, false, false);
  }

  // C/D layout: VGPR v -> row (tile_i + v + 8*lh), col (tile_j + lm)
  // Denominators are ~sqrt(768); v_rcp_f32 (1 ulp) instead of IEEE divide.
  const float qn0 = qn[b * Ln + j0 + lm];
  const float qn1 = qn[b * Ln + j0 + 16 + lm];
  const size_t rowbase = (size_t)b * Ln + i0 + 8 * lh;
  #pragma unroll
  for (int v = 0; v < 8; ++v) {
    float vn0 = vn[b * Ln + i0 + v + 8 * lh];
    float vn1 = vn[b * Ln + i0 + 16 + v + 8 * lh];
    size_t r0 = (rowbase + v) * Ln;
    size_t r1 = (rowbase + 16 + v) * Ln;
    SIM[r0 + j0 + lm]      = c00[v] * __builtin_amdgcn_rcpf(fmaxf(vn0 * qn0, 1e-8f));
    SIM[r0 + j0 + 16 + lm] = c01[v] * __builtin_amdgcn_rcpf(fmaxf(vn0 * qn1, 1e-8f));
    SIM[r1 + j0 + lm]      = c10[v] * __builtin_amdgcn_rcpf(fmaxf(vn1 * qn0, 1e-8f));
    SIM[r1 + j0 + 16 + lm] = c11[v] * __builtin_amdgcn_rcpf(fmaxf(vn1 * qn1, 1e-8f));
  }
}

// ---- Sinkhorn (diagonal-scaling form) ---------------------------------
__global__ __launch_bounds__(256) void k_sk_total(const float* __restrict__ SIM, float* T) {
  __shared__ float sh[256];
  int b = blockIdx.x, t = threadIdx.x;
  const float* p = SIM + (size_t)b * Ln * Ln;
  float acc = 0.f;
  for (int k = t; k < Ln * Ln; k += 256) acc += __expf(20.f * (p[k] - 1.f));
  sh[t] = acc; __syncthreads();
  for (int s = 128; s; s >>= 1) { if (t < s) sh[t] += sh[t + s]; __syncthreads(); }
  if (!t) T[b] = sh[0];
}

__global__ __launch_bounds__(256) void k_rc_init(const float* T, float* r, float* cv) {
  int idx = blockIdx.x * 256 + threadIdx.x;  // BL threads
  if (idx >= BL) return;
  r[idx]  = 1.f / T[idx >> 9];
  cv[idx] = 1.f;
}

__global__ __launch_bounds__(256) void k_sk_col(const float* __restrict__ SIM,
                                                const float* __restrict__ r, float* cv) {
  int b = blockIdx.x >> 1;
  int j = ((blockIdx.x & 1) << 8) + threadIdx.x;
  const float* p = SIM + (size_t)b * Ln * Ln + j;
  float acc = 0.f;
  for (int i = 0; i < Ln; ++i)
    acc += __expf(20.f * (p[(size_t)i * Ln] - 1.f)) * r[b * Ln + i];
  cv[b * Ln + j] = 1.f / ((float)Ln * acc);
}

__global__ __launch_bounds__(256) void k_sk_row(const float* __restrict__ SIM,
                                                const float* __restrict__ cv, float* r) {
  int row = blockIdx.x * 8 + (threadIdx.x >> 5);
  int lane = threadIdx.x & 31;
  int b = row >> 9;
  const float* p = SIM + (size_t)row * Ln;
  float acc = 0.f;
  for (int j = lane; j < Ln; j += 32)
    acc += __expf(20.f * (p[j] - 1.f)) * cv[b * Ln + j];
  acc = wred(acc);
  if (!lane) r[row] = 1.f / ((float)Ln * acc);
}

// ---- loss3 + row denominators (wave per row) --------------------------
__global__ __launch_bounds__(256) void k_loss3_denom(const float* __restrict__ SIM,
                                                     const float* __restrict__ r,
                                                     const float* __restrict__ cv,
                                                     const int* __restrict__ plen,
                                                     const int* __restrict__ tplen,
                                                     float* rowdenom, float* pla, float* plb) {
  __shared__ float sha[8], shb[8];
  int wid = threadIdx.x >> 5, lane = threadIdx.x & 31;
  int row = blockIdx.x * 8 + wid;
  int b = row >> 9, i = row & 511;
  float ri = r[row];
  float rowok = (i < tplen[b]) ? 1.f : 0.f;
  int pl = plen[b];
  const float* p = SIM + (size_t)row * Ln;
  float d = 0.f, la = 0.f, lb = 0.f;
  for (int j = lane; j < Ln; j += 32) {
    float s = p[j];
    float q = __expf(20.f * (s - 1.f));
    float P = (q * ri * cv[b * Ln + j] * (float)Ln >= 0.3f) ? 1.f : 0.f;
    float m = rowok * ((j < pl) ? 1.f : 0.f);
    d  += fmaxf(s, 0.f) * m + 0.001f;
    la += -P * logsig(s);
    lb += -(1.f - P) * logsig(-s);
  }
  d = wred(d); la = wred(la); lb = wred(lb);
  if (!lane) { rowdenom[row] = d; sha[wid] = la; shb[wid] = lb; }
  __syncthreads();
  if (threadIdx.x == 0) {
    float A = 0.f, Bv = 0.f;
    for (int k = 0; k < 8; ++k) { A += sha[k]; Bv += shb[k]; }
    pla[blockIdx.x] = A; plb[blockIdx.x] = Bv;
  }
}

// ---- contract j first: wt[b,i]=sum_j W*tm, wa=sum_j W*am --------------
__global__ __launch_bounds__(256) void k_wt(const float* __restrict__ SIM,
                                            const float* __restrict__ rowdenom,
                                            const int* __restrict__ plen,
                                            const int* __restrict__ tplen,
                                            const int* __restrict__ tm,
                                            const int* __restrict__ am,
                                            float* wt, float* wa) {
  int row = blockIdx.x * 8 + (threadIdx.x >> 5);
  int lane = threadIdx.x & 31;
  int b = row >> 9, i = row & 511;
  float rowok = (i < tplen[b]) ? 1.f : 0.f;
  int pl = plen[b];
  float dinv = 1.f / rowdenom[row];
  const float* p = SIM + (size_t)row * Ln;
  float at = 0.f, aa = 0.f;
  for (int j = lane; j < Ln; j += 32) {
    float s = p[j];
    float m = rowok * ((j < pl) ? 1.f : 0.f);
    float w = (fmaxf(s, 0.f) * m + 0.001f) * dinv * m;
    at += w * (float)tm[b * Ln + j];
    aa += w * (float)am[b * Ln + j];
  }
  at = wred(at); aa = wred(aa);
  if (!lane) { wt[row] = at; wa[row] = aa; }
}

// ---- t_trig[b,h] = sum_i wt[b,i]*V[b,i,h]/tcnt ------------------------
__global__ __launch_bounds__(256) void k_tmean(const float* __restrict__ V,
                                               const float* __restrict__ wt,
                                               const float* __restrict__ wa,
                                               const float* tcnt, const float* acnt,
                                               float* ttrig, float* targ) {
  int idx = blockIdx.x * 256 + threadIdx.x;
  if (idx >= BH) return;
  int b = idx / Hn, h = idx % Hn;
  const float* p = V + (size_t)b * Ln * Hn + h;
  float s1 = 0.f, s2 = 0.f;
  for (int i = 0; i < Ln; ++i) {
    float x = p[(size_t)i * Hn];
    s1 += wt[b * Ln + i] * x;
    s2 += wa[b * Ln + i] * x;
  }
  ttrig[idx] = s1 / tcnt[b];
  targ[idx]  = s2 / acnt[b];
}

// ---- FC + cross entropy for both feature sets -------------------------
__global__ __launch_bounds__(64) void k_fc_ce(const float* __restrict__ fc_w,
                                              const float* __restrict__ fc_b,
                                              const float* trig, const float* arg,
                                              const float* ttrig, const float* targ,
                                              const float* eemb, const float* aemb,
                                              const float* pemb,
                                              const int* etype, const int* atype,
                                              const int* rpos, const int* role,
                                              float* closs) {
  __shared__ float s1[NROLE], s2[NROLE];
  int b = blockIdx.x, rr = threadIdx.x;
  const float* w = fc_w + rr * FCIN;
  int et = etype[b], at = atype[b], rp = rpos[b];
  float a1 = fc_b[rr], a2 = fc_b[rr];
  for (int k = 0; k < Hn; ++k) { float wv = w[k];        a1 += wv * trig[b*Hn+k]; a2 += wv * ttrig[b*Hn+k]; }
  for (int k = 0; k < Hn; ++k) { float wv = w[Hn + k];   a1 += wv * arg[b*Hn+k];  a2 += wv * targ[b*Hn+k]; }
  for (int k = 0; k < En; ++k) { float wv = w[1536 + k]; float e = eemb[et*En+k]; a1 += wv*e; a2 += wv*e; }
  for (int k = 0; k < En; ++k) { float wv = w[1664 + k]; float e = aemb[at*En+k]; a1 += wv*e; a2 += wv*e; }
  for (int k = 0; k < En; ++k) { float wv = w[1792 + k]; float e = pemb[rp*En+k]; a1 += wv*e; a2 += wv*e; }
  s1[rr] = a1; s2[rr] = a2;
  __syncthreads();
  if (rr == 0) {
    int rl = role[b];
    float m1 = -1e30f, m2 = -1e30f;
    for (int k = 0; k < NROLE; ++k) { m1 = fmaxf(m1, s1[k]); m2 = fmaxf(m2, s2[k]); }
    float e1 = 0.f, e2 = 0.f;
    for (int k = 0; k < NROLE; ++k) { e1 += __expf(s1[k] - m1); e2 += __expf(s2[k] - m2); }
    float lse1 = m1 + __logf(e1), lse2 = m2 + __logf(e2);
    closs[b] = (lse1 - s1[rl]) + (lse2 - s2[rl]);
  }
}

// ---- final deterministic reduction ------------------------------------
__global__ __launch_bounds__(256) void k_final(const float* pla, const float* plb, int np,
                                               const float* closs, float* out) {
  __shared__ float sa[256], sb[256], sc[256];
  int t = threadIdx.x;
  float a = 0.f, b = 0.f, c = 0.f;
  for (int k = t; k < np; k += 256) { a += pla[k]; b += plb[k]; }
  if (t < Bn) c = closs[t];
  sa[t] = a; sb[t] = b; sc[t] = c;
  __syncthreads();
  for (int s = 128; s; s >>= 1) {
    if (t < s) { sa[t] += sa[t+s]; sb[t] += sb[t+s]; sc[t] += sc[t+s]; }
    __syncthreads();
  }
  if (!t) {
    const float inv = 1.f / (float)BLL;
    float loss3 = sa[0] * inv + sb[0] * inv;
    out[0] = sc[0] / (float)Bn + 0.5f * loss3;
  }
}

extern "C" void kernel_launch(void* const* d_in, const int* in_sizes, int n_in,
                              void* d_out, int out_size, void* d_ws, size_t ws_size,
                              hipStream_t stream) {
  const float* Q    = (const float*)d_in[0];   // transformer_output (queries)
  const float* V    = (const float*)d_in[1];   // trans_transformer_output (values)
  const float* fc_w = (const float*)d_in[2];
  const float* fc_b = (const float*)d_in[3];
  const float* eemb = (const float*)d_in[4];
  const float* aemb = (const float*)d_in[5];
  const float* pemb = (const float*)d_in[6];
  const int* tm     = (const int*)d_in[7];
  const int* am     = (const int*)d_in[8];
  const int* plen   = (const int*)d_in[9];     // pad_seq_len (queries_len)
  const int* tplen  = (const int*)d_in[10];    // trans_pad_seq_len (values_len)
  const int* etype  = (const int*)d_in[11];
  const int* atype  = (const int*)d_in[12];
  const int* rpos   = (const int*)d_in[13];
  const int* role   = (const int*)d_in[14];
  float* out = (float*)d_out;

  float* w = (float*)d_ws;
  float* SIM   = w;               w += BLL;     // 16777216
  float* trig  = w;               w += BH;
  float* arg   = w;               w += BH;
  float* ttrig = w;               w += BH;
  float* targ  = w;               w += BH;
  float* vn    = w;               w += BL;
  float* qn    = w;               w += BL;
  float* rv    = w;               w += BL;
  float* cv    = w;               w += BL;
  float* rowd  = w;               w += BL;
  float* wt    = w;               w += BL;
  float* wa    = w;               w += BL;
  float* T     = w;               w += Bn;
  float* tcnt  = w;               w += Bn;
  float* acnt  = w;               w += Bn;
  float* closs = w;               w += Bn;
  const int NP = 4096;
  float* pla   = w;               w += NP;
  float* plb   = w;               w += NP;

  k_maskcnt<<<8, 256, 0, stream>>>(tm, am, tcnt, acnt);
  k_masked_mean<<<BH / 256, 256, 0, stream>>>(Q, tm, am, tcnt, acnt, trig, arg);
  k_rownorm<<<2 * BL / 8, 256, 0, stream>>>(V, Q, vn, qn);
  k_gemm_sim<<<(Bn * 256) / 8, 256, 0, stream>>>(V, Q, vn, qn, SIM);

  k_sk_total<<<Bn, 256, 0, stream>>>(SIM, T);
  k_rc_init<<<BL / 256, 256, 0, stream>>>(T, rv, cv);
  for (int it = 0; it < 3; ++it) {
    k_sk_col<<<2 * Bn, 256, 0, stream>>>(SIM, rv, cv);
    k_sk_row<<<BL / 8, 256, 0, stream>>>(SIM, cv, rv);
  }

  k_loss3_denom<<<BL / 8, 256, 0, stream>>>(SIM, rv, cv, plen, tplen, rowd, pla, plb);
  k_wt<<<BL / 8, 256, 0, stream>>>(SIM, rowd, plen, tplen, tm, am, wt, wa);
  k_tmean<<<BH / 256, 256, 0, stream>>>(V, wt, wa, tcnt, acnt, ttrig, targ);
  k_fc_ce<<<Bn, NROLE, 0, stream>>>(fc_w, fc_b, trig, arg, ttrig, targ,
                                    eemb, aemb, pemb, etype, atype, rpos, role, closs);
  k_final<<<1, 256, 0, stream>>>(pla, plb, NP, closs, out);
}